// Policy_38147899523171
// MI455X (gfx1250) — compile-verified
//
#include <hip/hip_runtime.h>

// CDNA5 (gfx1250) GCN link-prediction pipeline, fp32 throughout.
// - V_WMMA_F32_16X16X4_F32 for the per-node 4x4 feature transform (exact fp32).
// - GCN norm factored out of the edge loop: node features pre-scaled by dinv,
//   accumulators seeded with the self-loop term, dinv[d] applied per-node.
// - Non-temporal loads/stores for the streaming edge list / logits so the
//   ~36 MB node working set stays resident in the 192 MB L2.
// - unsafeAtomicAdd -> hardware GLOBAL_ATOMIC_ADD_F32 (no CAS loops).

typedef float v2f __attribute__((ext_vector_type(2)));
typedef float v8f __attribute__((ext_vector_type(8)));

__device__ __forceinline__ void atomAddF(float* p, float v) {
#if defined(__HIP_DEVICE_COMPILE__)
  unsafeAtomicAdd(p, v);   // hardware GLOBAL_ATOMIC_ADD_F32
#else
  atomicAdd(p, v);
#endif
}

__device__ __forceinline__ int ntLoadI(const int* p) {
#if defined(__HIP_DEVICE_COMPILE__)
  return __builtin_nontemporal_load(p);
#else
  return *p;
#endif
}

// ---------------- Kernel 1: init degrees with self-loop ----------------
__global__ void k_init_deg(float* __restrict__ deg, int n) {
  int i = blockIdx.x * blockDim.x + threadIdx.x;
  if (i < n) deg[i] = 1.0f;
}

// ---------------- Kernel 2: degree count over dst ----------------
__global__ void k_count_deg(const int* __restrict__ dst, float* __restrict__ deg, int e) {
  int i = blockIdx.x * blockDim.x + threadIdx.x;
  if (i < e) atomAddF(&deg[ntLoadI(dst + i)], 1.0f);
}

// ---- Kernel 3: dinv = rsqrt(deg); hxs = x*W1*dinv; seed agg1 = hxs (self term) ----
__global__ void k_node_prep(const float* __restrict__ x, const float* __restrict__ W1,
                            float* __restrict__ dinv, float* __restrict__ hxs,
                            float* __restrict__ agg1, int n) {
  int i = blockIdx.x * blockDim.x + threadIdx.x;
  if (i >= n) return;
  float di = rsqrtf(dinv[i]);   // deg >= 1 always (self-loop)
  dinv[i] = di;
  float xd = x[i] * di;
  float4 h = make_float4(xd * W1[0], xd * W1[1], xd * W1[2], xd * W1[3]);
  ((float4*)hxs)[i] = h;
  ((float4*)agg1)[i] = h;       // self-loop seed: dinv[d] applied later per-node
}

// ---------------- Kernel 4/6: edge scatter  agg[d] += msg[s]  ----------------
// (msg is already scaled by dinv[s]; dinv[d] is applied in the node fixup.)
__global__ void k_edge_scatter(const int* __restrict__ src, const int* __restrict__ dst,
                               const float* __restrict__ msg, float* __restrict__ agg,
                               int e) {
  int i = blockIdx.x * blockDim.x + threadIdx.x;
  if (i >= e) return;
  int s = ntLoadI(src + i);
  int d = ntLoadI(dst + i);
  float4 m = ((const float4*)msg)[s];   // L2-resident gather
  float* a = agg + 4 * (size_t)d;
  atomAddF(a + 0, m.x);
  atomAddF(a + 1, m.y);
  atomAddF(a + 2, m.z);
  atomAddF(a + 3, m.w);
}

// ---- Kernel 5: h1 = relu(dinv*agg1 + b1); th = h1@W2 (WMMA fp32);
//      ths = th*dinv -> bufA; seed agg2 = ths (self term) -> bufB ----
// One wave handles 16 nodes with a single V_WMMA_F32_16X16X4_F32.
// NOTE: agg1 and agg2 alias (same buffer) -> no __restrict__ on them.
__global__ void k_transform_wmma(const float* agg1, const float* __restrict__ b1,
                                 const float* __restrict__ W2, const float* __restrict__ dinv,
                                 float* __restrict__ ths, float* agg2, int n) {
  int lane = threadIdx.x & 31;
  int wave = blockIdx.x * (blockDim.x >> 5) + (threadIdx.x >> 5);
  int half = lane >> 4;      // 0: K=0,1   1: K=2,3  (32-bit A 16x4 layout)
  int ml   = lane & 15;

#if defined(__AMDGCN__) && __has_builtin(__builtin_amdgcn_wmma_f32_16x16x4_f32)
  int base = __builtin_amdgcn_readfirstlane(wave * 16);  // SGPR -> uniform branches
  int node = base + ml;
  int nodeC = (node < n) ? node : (n - 1);   // clamp: keep EXEC all-1s for WMMA

  // A (16x4 fp32): lane L<16 holds {A[L][0],A[L][1]}, lane L+16 holds {A[L][2],A[L][3]}
  float2 av = ((const float2*)agg1)[nodeC * 2 + half];
  float din = dinv[nodeC];
  v2f A;
  A[0] = fmaxf(fmaf(din, av.x, b1[half * 2 + 0]), 0.0f);
  A[1] = fmaxf(fmaf(din, av.y, b1[half * 2 + 1]), 0.0f);
  // B (4x16 fp32): lane n<16 holds {B[0][n],B[1][n]}, lane n+16 holds {B[2][n],B[3][n]}
  // W2 is 4x4 row-major; columns 4..15 are zero-padding. Unconditional clamped
  // load + select avoids exec-mask branches before the WMMA.
  bool valid = (ml < 4);
  float w0 = W2[(half * 2 + 0) * 4 + (ml & 3)];
  float w1 = W2[(half * 2 + 1) * 4 + (ml & 3)];
  v2f B;
  B[0] = valid ? w0 : 0.0f;
  B[1] = valid ? w1 : 0.0f;
  v8f C = {};
  v8f D = __builtin_amdgcn_wmma_f32_16x16x4_f32(
      /*neg_a=*/false, A, /*neg_b=*/false, B,
      /*c_mod=*/(short)0, C, /*reuse_a=*/false, /*reuse_b=*/false);
  // D layout: VGPR r, lanes 0-15 -> (M=r, N=lane); lanes 16-31 -> (M=8+r, N=lane-16).
  // Lanes with (lane&15) < 4 hold valid output columns (features 0..3).
  if (base + 16 <= n) {                       // uniform fast path: full tile
    if (valid) {
      int mbase = base + half * 8;
#pragma unroll
      for (int r = 0; r < 8; ++r) {
        int m = mbase + r;
        float t = D[r] * dinv[m];             // ths = th * dinv
        ths[4 * m + ml]  = t;
        agg2[4 * m + ml] = t;                 // self-loop seed
      }
    }
  } else if (valid) {                         // tail tile: per-element guard
    int mbase = base + half * 8;
#pragma unroll
    for (int r = 0; r < 8; ++r) {
      int m = mbase + r;
      if (m < n) {
        float t = D[r] * dinv[m];
        ths[4 * m + ml]  = t;
        agg2[4 * m + ml] = t;
      }
    }
  }
#else
  // Scalar fallback (also used for the host compilation pass).
  int node = wave * 16 + ml;
  if (half == 0 && node < n) {
    float4 a = ((const float4*)agg1)[node];
    float din = dinv[node];
    float h0 = fmaxf(fmaf(din, a.x, b1[0]), 0.0f);
    float h1v = fmaxf(fmaf(din, a.y, b1[1]), 0.0f);
    float h2v = fmaxf(fmaf(din, a.z, b1[2]), 0.0f);
    float h3v = fmaxf(fmaf(din, a.w, b1[3]), 0.0f);
#pragma unroll
    for (int j = 0; j < 4; ++j) {
      float t = (h0 * W2[0 * 4 + j] + h1v * W2[1 * 4 + j] +
                 h2v * W2[2 * 4 + j] + h3v * W2[3 * 4 + j]) * din;
      ths[4 * node + j]  = t;
      agg2[4 * node + j] = t;
    }
  }
#endif
}

// ---------------- Kernel 7: h2 = dinv*agg2 + b2 (in place) ----------------
__global__ void k_finalize(float* __restrict__ h2, const float* __restrict__ dinv,
                           const float* __restrict__ b2, int n) {
  int i = blockIdx.x * blockDim.x + threadIdx.x;
  if (i >= n) return;
  float di = dinv[i];
  float4 v = ((float4*)h2)[i];
  v.x = fmaf(di, v.x, b2[0]);
  v.y = fmaf(di, v.y, b2[1]);
  v.z = fmaf(di, v.z, b2[2]);
  v.w = fmaf(di, v.w, b2[3]);
  ((float4*)h2)[i] = v;
}

// ---------------- Kernel 8: logits[e] = dot(h2[src], h2[dst]) ----------------
__global__ void k_score(const int* __restrict__ src, const int* __restrict__ dst,
                        const float* __restrict__ h2, float* __restrict__ out, int e) {
  int i = blockIdx.x * blockDim.x + threadIdx.x;
  if (i >= e) return;
  float4 a = ((const float4*)h2)[ntLoadI(src + i)];
  float4 b = ((const float4*)h2)[ntLoadI(dst + i)];
  float r = a.x * b.x + a.y * b.y + a.z * b.z + a.w * b.w;
#if defined(__HIP_DEVICE_COMPILE__)
  __builtin_nontemporal_store(r, out + i);   // pure stream, keep L2 for nodes
#else
  out[i] = r;
#endif
}

extern "C" void kernel_launch(void* const* d_in, const int* in_sizes, int n_in,
                              void* d_out, int out_size, void* d_ws, size_t ws_size,
                              hipStream_t stream) {
  const float* x  = (const float*)d_in[0];
  const int*   ei = (const int*)d_in[1];
  const float* W1 = (const float*)d_in[2];
  const float* b1 = (const float*)d_in[3];
  const float* W2 = (const float*)d_in[4];
  const float* b2 = (const float*)d_in[5];
  float* out = (float*)d_out;

  int n = in_sizes[0];       // N_NODES (x is [N,1])
  int e = in_sizes[1] / 2;   // edge_index is [2,E]
  const int* src = ei;
  const int* dst = ei + e;

  // Workspace (floats): dinv[N] | bufA[4N] (hxs -> ths) | bufB[4N] (agg1 -> agg2 -> h2)
  float* dinv = (float*)d_ws;
  float* bufA = dinv + (size_t)n;
  float* bufB = bufA + 4 * (size_t)n;

  int gn = (n + 255) / 256;
  int ge = (e + 255) / 256;
  int gw = (n + 127) / 128;  // WMMA kernel: 256 thr = 8 waves = 128 nodes/block

  k_init_deg      <<<gn, 256, 0, stream>>>(dinv, n);
  k_count_deg     <<<ge, 256, 0, stream>>>(dst, dinv, e);
  k_node_prep     <<<gn, 256, 0, stream>>>(x, W1, dinv, bufA, bufB, n);
  k_edge_scatter  <<<ge, 256, 0, stream>>>(src, dst, bufA, bufB, e);        // layer 1 agg
  k_transform_wmma<<<gw, 256, 0, stream>>>(bufB, b1, W2, dinv, bufA, bufB, n);
  k_edge_scatter  <<<ge, 256, 0, stream>>>(src, dst, bufA, bufB, e);        // layer 2 agg
  k_finalize      <<<gn, 256, 0, stream>>>(bufB, dinv, b2, n);
  k_score         <<<ge, 256, 0, stream>>>(src, dst, bufB, out, e);
}